// NormalizeWordsModel_70162585747874
// MI455X (gfx1250) — compile-verified
//
#include <hip/hip_runtime.h>
#include <hip/hip_bf16.h>
#include <stdint.h>

// Problem constants (reference: T=8192, V=50000, D=300)
#define T_    8192
#define V_    50000
#define D_    300
#define DP    308          // padded LDS row pitch in floats (16B-aligned rows, K-pad zeroed)
#define KSTEPS 76          // 304 / 4  (K padded to 304 for 16x16x4 f32 WMMA)
#define NTILES 3125        // V_ / 16
#define WAVES  4
#define THREADS (WAVES*32)
#define QPW   16           // queries per wave (one 16x16 WMMA N-tile)
#define QPB   (WAVES*QPW)  // 64 queries per block
#define DCH   19           // ceil(304/16) d-chunks of 16

typedef float v2f __attribute__((ext_vector_type(2)));
typedef float v8f __attribute__((ext_vector_type(8)));

// ---------------------------------------------------------------------------
// Kernel 1: q = words @ W   (T x D) @ (D x D) -> workspace
// ---------------------------------------------------------------------------
__global__ __launch_bounds__(256) void qproj_kernel(const float* __restrict__ words,
                                                    const float* __restrict__ W,
                                                    float* __restrict__ q) {
    __shared__ float wrow[D_];
    const int t = blockIdx.x;
    for (int c = threadIdx.x; c < D_; c += 256) wrow[c] = words[(size_t)t * D_ + c];
    __syncthreads();
    for (int d = threadIdx.x; d < D_; d += 256) {
        float acc = 0.f;
#pragma unroll 4
        for (int k = 0; k < D_; ++k) acc = fmaf(wrow[k], W[k * D_ + d], acc);
        q[(size_t)t * D_ + d] = acc;
    }
}

// ---------------------------------------------------------------------------
// Kernel 2: streaming softmax-attention over the vocab with fp32 WMMA.
// Each wave owns 16 queries.  Per 16-row vocab tile:
//   S^T[v][m] = vocab_tile @ q^T       (76 x v_wmma_f32_16x16x4_f32)
//   online softmax per lane (query m in lane dim), P staged via LDS
//   acc[m][d] += P @ vocab_tile        (19*4 x v_wmma_f32_16x16x4_f32)
// Vocab tiles staged to LDS with async b128 copies, double buffered.
// ---------------------------------------------------------------------------
__global__ __launch_bounds__(THREADS) void flash_kernel(const float* __restrict__ words,
                                                        const float* __restrict__ vocab,
                                                        const float* __restrict__ de,
                                                        const float* __restrict__ q,
                                                        float* __restrict__ out) {
    __shared__ __align__(16) float qb[QPB * DP];        // 78.8 KB  q tile (zero K-pad)
    __shared__ __align__(16) float vt[2][16 * DP];      // 39.4 KB  vocab tile double buffer
    __shared__ __align__(16) float pb[WAVES][QPW * 16]; //  4 KB    P re-layout staging
    __shared__ __align__(16) float sb[WAVES][16];       // rescale broadcast
    __shared__ __align__(16) float2 sb2[WAVES][16];     // epilogue broadcast

    const int tid  = threadIdx.x;
    const int wave = tid >> 5;
    const int lane = tid & 31;
    const int qm   = lane & 15;  // row index inside 16-tile (A row / B col / query)
    const int half = lane >> 4;  // K-half select for fragments
    const int qbase = blockIdx.x * QPB;
    const float* qw = q + (size_t)qbase * D_;

    // ---- zero K-pad columns, stage q tile to LDS ----
    for (int e = tid; e < QPB * 8; e += THREADS) {
        int r = e >> 3, c = (e & 7) + D_;
        qb[r * DP + c] = 0.f;
    }
    for (int e = tid; e < 2 * 16 * 8; e += THREADS) {
        int b = e >> 7, rem = e & 127;
        vt[b][(rem >> 3) * DP + (rem & 7) + D_] = 0.f;
    }
    for (int e = tid; e < QPB * D_; e += THREADS) {
        int r = e / D_, c = e % D_;
        qb[r * DP + c] = qw[(size_t)r * D_ + c];
    }
    __syncthreads();

    // ---- l_def[m] = q[m] . default_embed (split across lane halves) ----
    float part = 0.f;
    {
        const float* qrow = &qb[(wave * QPW + qm) * DP];
        const int k0 = half * 150;
        for (int i = 0; i < 150; ++i) part = fmaf(qrow[k0 + i], de[k0 + i], part);
    }
    const float ldef = part + __shfl_xor(part, 16);

    float mrun = -3.0e38f, lsum = 0.f;
    const v8f vzero = {0.f, 0.f, 0.f, 0.f, 0.f, 0.f, 0.f, 0.f};
    v8f acc[DCH];
#pragma unroll
    for (int j = 0; j < DCH; ++j) acc[j] = vzero;

    // ---- async vocab tile copy: 16 rows x 300 floats = 1200 b128 units ----
    auto issue_copy = [&](int buf, int v0) {
        const uint32_t ldsb = (uint32_t)(uintptr_t)&vt[buf][0];
#pragma unroll
        for (int i = 0; i < 10; ++i) {
            int u = tid + THREADS * i;
            if (u < 16 * 75) {
                int r = u / 75, c = u % 75;
                uint32_t lo = ldsb + (uint32_t)(r * DP * 4 + c * 16);
                uint32_t go = (uint32_t)((v0 + r) * (D_ * 4) + c * 16);
                asm volatile("global_load_async_to_lds_b128 %0, %1, %2"
                             :: "v"(lo), "v"(go), "s"(vocab) : "memory");
            }
        }
    };

    issue_copy(0, 0);
    int cur = 0;

    const float* qrow = &qb[(wave * QPW + qm) * DP];

    for (int tile = 0; tile < NTILES; ++tile) {
        asm volatile("s_wait_asynccnt 0" ::: "memory");
        __syncthreads();
        if (tile + 1 < NTILES) issue_copy(cur ^ 1, (tile + 1) * 16);

        const float* vtc  = &vt[cur][0];
        const float* vrow = &vtc[qm * DP];

        // ---- S^T = vocab_tile(16xK) @ q^T(Kx16), two interleaved accumulators
        v8f c8a = vzero, c8b = vzero;
#pragma unroll 4
        for (int s = 0; s < KSTEPS; s += 2) {
            v2f av0 = *(const v2f*)(vrow + 4 * s + 2 * half);
            v2f bq0 = *(const v2f*)(qrow + 4 * s + 2 * half);
            c8a = __builtin_amdgcn_wmma_f32_16x16x4_f32(false, av0, false, bq0,
                                                        (short)0, c8a, false, false);
            v2f av1 = *(const v2f*)(vrow + 4 * (s + 1) + 2 * half);
            v2f bq1 = *(const v2f*)(qrow + 4 * (s + 1) + 2 * half);
            c8b = __builtin_amdgcn_wmma_f32_16x16x4_f32(false, av1, false, bq1,
                                                        (short)0, c8b, false, false);
        }
        v8f c8 = c8a + c8b;

        // ---- online softmax: lane = (query qm, v-rows 8*half..8*half+7) ----
        float tmax = fmaxf(fmaxf(fmaxf(c8[0], c8[1]), fmaxf(c8[2], c8[3])),
                           fmaxf(fmaxf(c8[4], c8[5]), fmaxf(c8[6], c8[7])));
        tmax = fmaxf(tmax, __shfl_xor(tmax, 16));
        const float Mn  = fmaxf(mrun, tmax);
        const float scl = __expf(mrun - Mn);
        mrun = Mn;
        float pr[8], ts = 0.f;
#pragma unroll
        for (int r = 0; r < 8; ++r) { pr[r] = __expf(c8[r] - Mn); ts += pr[r]; }
        lsum = lsum * scl + ts;

        // ---- stage P into A-fragment layout: pb[m*16 + k] = P[m][k] ----
        float4* pdst = (float4*)&pb[wave][qm * 16 + half * 8];
        pdst[0] = make_float4(pr[0], pr[1], pr[2], pr[3]);
        pdst[1] = make_float4(pr[4], pr[5], pr[6], pr[7]);
        if (lane < 16) sb[wave][qm] = scl;

        float2 af[4];
#pragma unroll
        for (int s = 0; s < 4; ++s)
            af[s] = *(const float2*)&pb[wave][qm * 16 + 4 * s + 2 * half];
        const float4 sA = *(const float4*)&sb[wave][half * 8];
        const float4 sB = *(const float4*)&sb[wave][half * 8 + 4];

        // ---- acc[m][d] = acc*scale + P(16x16) @ vocab_tile(16xD) ----
#pragma unroll
        for (int j = 0; j < DCH; ++j) {
            v8f a = acc[j];
            a[0] *= sA.x; a[1] *= sA.y; a[2] *= sA.z; a[3] *= sA.w;
            a[4] *= sB.x; a[5] *= sB.y; a[6] *= sB.z; a[7] *= sB.w;
            const float* bcol = vtc + j * 16 + qm;
#pragma unroll
            for (int s = 0; s < 4; ++s) {
                const int k = 4 * s + 2 * half;
                v2f bfv = {bcol[k * DP], bcol[(k + 1) * DP]};
                v2f afv = {af[s].x, af[s].y};
                a = __builtin_amdgcn_wmma_f32_16x16x4_f32(false, afv, false, bfv,
                                                          (short)0, a, false, false);
            }
            acc[j] = a;
        }
        cur ^= 1;
    }

    // ---- epilogue: fold in default-embed column (value = words row) ----
    lsum += __shfl_xor(lsum, 16);
    const float ed  = __expf(ldef - mrun);
    const float inv = 1.0f / (lsum + ed);
    if (lane < 16) sb2[wave][qm] = make_float2(inv, ed * inv);
    float2 f8[8];
#pragma unroll
    for (int r = 0; r < 8; ++r) f8[r] = sb2[wave][half * 8 + r];

#pragma unroll
    for (int j = 0; j < DCH; ++j) {
        const int d = j * 16 + qm;
        if (d < D_) {
#pragma unroll
            for (int r = 0; r < 8; ++r) {
                const int gq = qbase + wave * QPW + half * 8 + r;
                const float wv = words[(size_t)gq * D_ + d];
                out[(size_t)gq * D_ + d] = acc[j][r] * f8[r].x + wv * f8[r].y;
            }
        }
    }
}

// ---------------------------------------------------------------------------
extern "C" void kernel_launch(void* const* d_in, const int* in_sizes, int n_in,
                              void* d_out, int out_size, void* d_ws, size_t ws_size,
                              hipStream_t stream) {
    const float* words = (const float*)d_in[0];  // [T, D]
    const float* vocab = (const float*)d_in[1];  // [V, D]
    const float* de    = (const float*)d_in[2];  // [D]
    const float* W     = (const float*)d_in[3];  // [D, D]
    float* out = (float*)d_out;                  // [T, D]
    float* qws = (float*)d_ws;                   // [T, D] scratch (9.83 MB)

    qproj_kernel<<<T_, 256, 0, stream>>>(words, W, qws);
    flash_kernel<<<T_ / QPB, THREADS, 0, stream>>>(words, vocab, de, qws, out);
}